// BiMamba2_28776280883416
// MI455X (gfx1250) — compile-verified
//
#include <hip/hip_runtime.h>
#include <math.h>

typedef float v2f __attribute__((ext_vector_type(2)));
typedef float v8f __attribute__((ext_vector_type(8)));

constexpr int  BATCH  = 2;
constexpr int  LSEQ   = 2048;
constexpr int  DMODEL = 1024;
constexpr int  DINNER = 2048;     // 2 * DMODEL
constexpr int  DSTATE = 16;
constexpr int  DTRANK = 64;
constexpr int  XDBLW  = DTRANK + 2 * DSTATE;   // 96
constexpr int  XPW_PAD= 128;                   // padded N for x_proj weights
constexpr long MROWS  = (long)BATCH * LSEQ;    // 4096

// ---------------------------------------------------------------------------
// One wave computes a 16x64 tile of C = A * W^T, K-loop software-pipelined
// (1-deep prefetch, unconditional float2 loads -> no exec-mask churn).
// Fragment layout for V_WMMA_F32_16X16X4_F32 (wave32):
//   A (16x4):  lanes 0-15 -> M, lane half selects K{0,2}; float2 at (m, k0+2h)
//   B (4x16):  mirrored on W rows: float2 at (n, k0+2h)
//   C (16x16): VGPR i -> M = i + 8*half, N = lane&15
// Caller guarantees W rows [col0, col0+64) are readable (pad if needed).
// ---------------------------------------------------------------------------
__device__ inline void wmma_gemm_16x64(const float* __restrict__ Abase,
                                       long rstrideA,
                                       const float* __restrict__ W,
                                       int ldw, int K, int col0,
                                       v8f c[4]) {
  const int lane = threadIdx.x & 31;
  const int half = lane >> 4;   // 0 or 1
  const int sub  = lane & 15;
  const float* ap  = Abase + (long)sub * rstrideA + 2 * half;
  const float* wp0 = W + (long)(col0      + sub) * ldw + 2 * half;
  const float* wp1 = W + (long)(col0 + 16 + sub) * ldw + 2 * half;
  const float* wp2 = W + (long)(col0 + 32 + sub) * ldw + 2 * half;
  const float* wp3 = W + (long)(col0 + 48 + sub) * ldw + 2 * half;

  float2 a0  = *(const float2*)(ap);
  float2 b00 = *(const float2*)(wp0);
  float2 b01 = *(const float2*)(wp1);
  float2 b02 = *(const float2*)(wp2);
  float2 b03 = *(const float2*)(wp3);

  for (int k0 = 4; k0 < K; k0 += 4) {
    // prefetch next k-step
    float2 a1  = *(const float2*)(ap  + k0);
    float2 b10 = *(const float2*)(wp0 + k0);
    float2 b11 = *(const float2*)(wp1 + k0);
    float2 b12 = *(const float2*)(wp2 + k0);
    float2 b13 = *(const float2*)(wp3 + k0);
    // compute on current fragments
    v2f a; a.x = a0.x; a.y = a0.y;
    v2f b;
    b.x = b00.x; b.y = b00.y;
    c[0] = __builtin_amdgcn_wmma_f32_16x16x4_f32(false, a, false, b, (short)0, c[0], false, false);
    b.x = b01.x; b.y = b01.y;
    c[1] = __builtin_amdgcn_wmma_f32_16x16x4_f32(false, a, false, b, (short)0, c[1], false, false);
    b.x = b02.x; b.y = b02.y;
    c[2] = __builtin_amdgcn_wmma_f32_16x16x4_f32(false, a, false, b, (short)0, c[2], false, false);
    b.x = b03.x; b.y = b03.y;
    c[3] = __builtin_amdgcn_wmma_f32_16x16x4_f32(false, a, false, b, (short)0, c[3], false, false);
    a0 = a1; b00 = b10; b01 = b11; b02 = b12; b03 = b13;
  }
  {
    v2f a; a.x = a0.x; a.y = a0.y;
    v2f b;
    b.x = b00.x; b.y = b00.y;
    c[0] = __builtin_amdgcn_wmma_f32_16x16x4_f32(false, a, false, b, (short)0, c[0], false, false);
    b.x = b01.x; b.y = b01.y;
    c[1] = __builtin_amdgcn_wmma_f32_16x16x4_f32(false, a, false, b, (short)0, c[1], false, false);
    b.x = b02.x; b.y = b02.y;
    c[2] = __builtin_amdgcn_wmma_f32_16x16x4_f32(false, a, false, b, (short)0, c[2], false, false);
    b.x = b03.x; b.y = b03.y;
    c[3] = __builtin_amdgcn_wmma_f32_16x16x4_f32(false, a, false, b, (short)0, c[3], false, false);
  }
}

__device__ inline void store_tile(float* __restrict__ C, long ldc,
                                  int row0, int col0, int Nvalid,
                                  const v8f c[4]) {
  const int lane = threadIdx.x & 31;
  const int half = lane >> 4;
  const int sub  = lane & 15;
#pragma unroll
  for (int j = 0; j < 4; ++j) {
    int col = col0 + 16 * j + sub;
    if (col >= Nvalid) continue;
#pragma unroll
    for (int i = 0; i < 8; ++i)
      C[(long)(row0 + i + 8 * half) * ldc + col] = c[j][i];
  }
}

// ---------------------------------------------------------------------------
// K0: zero-pad x_proj weights [96 x 2048] -> [128 x 2048] per direction
// ---------------------------------------------------------------------------
__global__ __launch_bounds__(256) void k_pad_xpw(const float* __restrict__ Wf,
                                                 const float* __restrict__ Wb,
                                                 float* __restrict__ Wpad) {
  const int i = blockIdx.x * 256 + threadIdx.x;         // 2*128*2048
  const int per = XPW_PAD * DINNER;
  if (i >= 2 * per) return;
  const int dir = i / per;
  const int rem = i - dir * per;
  const int n = rem / DINNER;
  const int k = rem - n * DINNER;
  const float* W = dir ? Wb : Wf;
  Wpad[i] = (n < XDBLW) ? W[n * DINNER + k] : 0.0f;
}

// ---------------------------------------------------------------------------
// K1: xz[dir] = x(flipped for dir==1) @ in_proj_w^T     [4096 x 4096]
// ---------------------------------------------------------------------------
__global__ __launch_bounds__(256) void k_in_proj(const float* __restrict__ x,
                                                 const float* __restrict__ Wf,
                                                 const float* __restrict__ Wb,
                                                 float* __restrict__ xz) {
  const int dir  = blockIdx.z;
  const float* W = dir ? Wb : Wf;
  const int wave = threadIdx.x >> 5;
  const int row0 = blockIdx.x * 128 + wave * 16;
  const int col0 = blockIdx.y * 64;
  const int bb = row0 >> 11;       // batch index (L = 2048)
  const int t0 = row0 & 2047;
  const long arow0   = dir ? ((long)bb * LSEQ + (LSEQ - 1 - t0)) : (long)row0;
  const long rstride = dir ? -(long)DMODEL : (long)DMODEL;
  v8f c[4] = {};
  wmma_gemm_16x64(x + arow0 * DMODEL, rstride, W, DMODEL, DMODEL, col0, c);
  store_tile(xz + (long)dir * MROWS * (2 * DINNER), 2 * DINNER, row0, col0, 2 * DINNER, c);
}

// ---------------------------------------------------------------------------
// K2: depthwise causal conv1d (k=4, left pad) + bias + SiLU -> xc (u)
// ---------------------------------------------------------------------------
__global__ __launch_bounds__(256) void k_conv_silu(const float* __restrict__ xz,
                                                   const float* __restrict__ cwf,
                                                   const float* __restrict__ cwb,
                                                   const float* __restrict__ cbf,
                                                   const float* __restrict__ cbb,
                                                   float* __restrict__ xc) {
  const long i = (long)blockIdx.x * blockDim.x + threadIdx.x;
  const long total = 2 * MROWS * (long)DINNER;
  if (i >= total) return;
  const int  d  = (int)(i & (DINNER - 1));
  const long rr = i >> 11;            // dir*MROWS + r
  const int  dir = (rr >= MROWS);
  const long r   = rr - (long)dir * MROWS;   // b*L + t
  const int  t   = (int)(r & (LSEQ - 1));
  const float* cw = (dir ? cwb : cwf) + d * 4;
  float acc = (dir ? cbb : cbf)[d];
  const float* base = xz + (long)dir * MROWS * (2 * DINNER);
#pragma unroll
  for (int j = 0; j < 4; ++j) {
    int tt = t - 3 + j;
    if (tt >= 0) acc += cw[j] * base[(r - 3 + j) * (2 * DINNER) + d];
  }
  xc[i] = acc / (1.0f + __expf(-acc));
}

// ---------------------------------------------------------------------------
// K3: x_dbl[dir] = xc[dir] @ x_proj_w^T   [4096 x 96] (padded W, masked store)
// ---------------------------------------------------------------------------
__global__ __launch_bounds__(256) void k_x_proj(const float* __restrict__ xc,
                                                const float* __restrict__ Wpad,
                                                float* __restrict__ xdbl) {
  const int dir  = blockIdx.z;
  const float* W = Wpad + (long)dir * XPW_PAD * DINNER;
  const int wave = threadIdx.x >> 5;
  const int row0 = blockIdx.x * 128 + wave * 16;
  const int col0 = blockIdx.y * 64;
  v8f c[4] = {};
  wmma_gemm_16x64(xc + ((long)dir * MROWS + row0) * DINNER, DINNER,
                  W, DINNER, DINNER, col0, c);
  store_tile(xdbl + (long)dir * MROWS * XDBLW, XDBLW, row0, col0, XDBLW, c);
}

// ---------------------------------------------------------------------------
// K4: dt[dir] = softplus(x_dbl[:, :64] @ dt_proj_w^T + dt_proj_b)  [4096 x 2048]
// ---------------------------------------------------------------------------
__global__ __launch_bounds__(256) void k_dt_proj(const float* __restrict__ xdbl,
                                                 const float* __restrict__ Wf,
                                                 const float* __restrict__ Wb,
                                                 const float* __restrict__ biasf,
                                                 const float* __restrict__ biasb,
                                                 float* __restrict__ dtout) {
  const int dir  = blockIdx.z;
  const float* W    = dir ? Wb : Wf;
  const float* bias = dir ? biasb : biasf;
  const int wave = threadIdx.x >> 5;
  const int row0 = blockIdx.x * 128 + wave * 16;
  const int col0 = blockIdx.y * 64;
  v8f c[4] = {};
  wmma_gemm_16x64(xdbl + ((long)dir * MROWS + row0) * XDBLW, XDBLW,
                  W, DTRANK, DTRANK, col0, c);
  float* out = dtout + (long)dir * MROWS * DINNER;
  const int lane = threadIdx.x & 31;
  const int half = lane >> 4;
  const int sub  = lane & 15;
#pragma unroll
  for (int j = 0; j < 4; ++j) {
    const int col = col0 + 16 * j + sub;
    const float bv = bias[col];
#pragma unroll
    for (int i = 0; i < 8; ++i) {
      float v = c[j][i] + bv;
      float sp = (v > 20.0f) ? v : log1pf(expf(v));
      out[(long)(row0 + i + 8 * half) * DINNER + col] = sp;
    }
  }
}

// ---------------------------------------------------------------------------
// K5: selective scan. One lane per (dir, b, d). 16-wide state in registers.
// Writes gated y (y * silu(z)) in-place over xc.
// ---------------------------------------------------------------------------
__global__ __launch_bounds__(256) void k_scan(const float* __restrict__ xz,
                                              const float* __restrict__ xdbl,
                                              const float* __restrict__ dtb,
                                              const float* __restrict__ Alog_f,
                                              const float* __restrict__ Alog_b,
                                              const float* __restrict__ Df,
                                              const float* __restrict__ Db,
                                              float* __restrict__ xcy) {
  const int gid = blockIdx.x * blockDim.x + threadIdx.x;  // 8192 total
  const int dir = gid >> 12;
  const int b   = (gid >> 11) & 1;
  const int d   = gid & (DINNER - 1);
  const float* Alog = (dir ? Alog_b : Alog_f) + d * DSTATE;
  float a[DSTATE];
#pragma unroll
  for (int n = 0; n < DSTATE; ++n) a[n] = -__expf(Alog[n]);
  const float Dv = (dir ? Db : Df)[d];
  float h[DSTATE] = {};
  const float* xdp   = xdbl + ((long)dir * MROWS + (long)b * LSEQ) * XDBLW;
  const float* zbase = xz + (long)dir * MROWS * (2 * DINNER)
                          + (long)b * LSEQ * (2 * DINNER) + DINNER + d;
  long idx = ((long)dir * MROWS + (long)b * LSEQ) * DINNER + d;
  for (int t = 0; t < LSEQ; ++t) {
    const float u   = xcy[idx];
    const float dtv = dtb[idx];
    const float* row = xdp + (long)t * XDBLW;
    const float dtu = dtv * u;
    float y = 0.f;
#pragma unroll
    for (int n = 0; n < DSTATE; ++n) {
      float hn = __expf(dtv * a[n]) * h[n] + dtu * row[DTRANK + n];
      h[n] = hn;
      y += hn * row[DTRANK + DSTATE + n];
    }
    y += Dv * u;
    const float z = zbase[(long)t * (2 * DINNER)];
    y *= z / (1.0f + __expf(-z));
    xcy[idx] = y;
    idx += DINNER;
  }
}

// ---------------------------------------------------------------------------
// K6: out[:, :, dir*1024 + n] = y_gated @ out_proj_w^T (bwd un-flips time)
// ---------------------------------------------------------------------------
__global__ __launch_bounds__(256) void k_out_proj(const float* __restrict__ y,
                                                  const float* __restrict__ Wf,
                                                  const float* __restrict__ Wb,
                                                  float* __restrict__ out) {
  const int dir  = blockIdx.z;
  const float* W = dir ? Wb : Wf;
  const int wave = threadIdx.x >> 5;
  const int row0 = blockIdx.x * 128 + wave * 16;
  const int col0 = blockIdx.y * 64;
  v8f c[4] = {};
  wmma_gemm_16x64(y + ((long)dir * MROWS + row0) * DINNER, DINNER,
                  W, DINNER, DINNER, col0, c);
  const int lane = threadIdx.x & 31;
  const int half = lane >> 4;
  const int sub  = lane & 15;
#pragma unroll
  for (int j = 0; j < 4; ++j) {
    const int col = col0 + 16 * j + sub;
#pragma unroll
    for (int i = 0; i < 8; ++i) {
      const int r = row0 + i + 8 * half;
      const int bb = r >> 11;
      const int t  = r & (LSEQ - 1);
      const int tout = dir ? (LSEQ - 1 - t) : t;
      out[((long)bb * LSEQ + tout) * (2 * DMODEL) + dir * DMODEL + col] = c[j][i];
    }
  }
}

// ---------------------------------------------------------------------------
extern "C" void kernel_launch(void* const* d_in, const int* in_sizes, int n_in,
                              void* d_out, int out_size, void* d_ws, size_t ws_size,
                              hipStream_t stream) {
  (void)in_sizes; (void)n_in; (void)out_size; (void)ws_size;
  const float* x        = (const float*)d_in[0];
  const float* f_inw    = (const float*)d_in[1];
  const float* f_convw  = (const float*)d_in[2];
  const float* f_convb  = (const float*)d_in[3];
  const float* f_xpw    = (const float*)d_in[4];
  const float* f_dtw    = (const float*)d_in[5];
  const float* f_dtb    = (const float*)d_in[6];
  const float* f_Alog   = (const float*)d_in[7];
  const float* f_D      = (const float*)d_in[8];
  const float* f_outw   = (const float*)d_in[9];
  const float* b_inw    = (const float*)d_in[10];
  const float* b_convw  = (const float*)d_in[11];
  const float* b_convb  = (const float*)d_in[12];
  const float* b_xpw    = (const float*)d_in[13];
  const float* b_dtw    = (const float*)d_in[14];
  const float* b_dtb    = (const float*)d_in[15];
  const float* b_Alog   = (const float*)d_in[16];
  const float* b_D      = (const float*)d_in[17];
  const float* b_outw   = (const float*)d_in[18];
  float* out = (float*)d_out;

  // Workspace carve-up (floats): xz | xc(u->y) | dt | xdbl | Wpad  (~274 MB)
  float* ws   = (float*)d_ws;
  float* xz   = ws;                                        // 2 * 4096 * 4096
  float* xc   = xz + 2 * MROWS * (2 * (long)DINNER);       // 2 * 4096 * 2048
  float* dtb  = xc + 2 * MROWS * (long)DINNER;             // 2 * 4096 * 2048
  float* xdbl = dtb + 2 * MROWS * (long)DINNER;            // 2 * 4096 * 96
  float* wpad = xdbl + 2 * MROWS * (long)XDBLW;            // 2 * 128 * 2048

  dim3 blk(256);
  // K0: pad x_proj weights so the GEMM needs no load guards
  {
    int total = 2 * XPW_PAD * DINNER;
    k_pad_xpw<<<dim3((total + 255) / 256), blk, 0, stream>>>(f_xpw, b_xpw, wpad);
  }
  // K1: in_proj GEMM  (M=4096, N=4096, K=1024), dirs in grid.z
  k_in_proj<<<dim3(MROWS / 128, (2 * DINNER) / 64, 2), blk, 0, stream>>>(x, f_inw, b_inw, xz);
  // K2: conv + silu
  {
    long total = 2 * MROWS * (long)DINNER;
    k_conv_silu<<<dim3((unsigned)((total + 255) / 256)), blk, 0, stream>>>(
        xz, f_convw, b_convw, f_convb, b_convb, xc);
  }
  // K3: x_proj GEMM (N=96 via padded 128)
  k_x_proj<<<dim3(MROWS / 128, XPW_PAD / 64, 2), blk, 0, stream>>>(xc, wpad, xdbl);
  // K4: dt_proj GEMM + softplus (N=2048, K=64)
  k_dt_proj<<<dim3(MROWS / 128, DINNER / 64, 2), blk, 0, stream>>>(
      xdbl, f_dtw, b_dtw, f_dtb, b_dtb, dtb);
  // K5: selective scan + gating (8192 lanes)
  k_scan<<<dim3(32), blk, 0, stream>>>(xz, xdbl, dtb, f_Alog, b_Alog, f_D, b_D, xc);
  // K6: out_proj GEMM -> concatenated output (N=1024)
  k_out_proj<<<dim3(MROWS / 128, DMODEL / 64, 2), blk, 0, stream>>>(xc, f_outw, b_outw, out);
}